// GCN_68178310856719
// MI455X (gfx1250) — compile-verified
//
#include <hip/hip_runtime.h>
#include <hip/hip_bf16.h>

// ---------------------------------------------------------------------------
// 2-layer GCN for gfx1250 (MI455X).
//   - fp32 WMMA (V_WMMA_F32_16X16X4_F32) for the two N x 64 x 64 GEMMs
//   - L2-resident atomic scatter-add for the edge aggregation (the bottleneck)
// ---------------------------------------------------------------------------

typedef __attribute__((ext_vector_type(2))) float v2f;
typedef __attribute__((ext_vector_type(8))) float v8f;

#define TB 256

// deg[i] = 1 (self loop)
__global__ void k_deg_init(float* __restrict__ deg, int n) {
  int i = blockIdx.x * blockDim.x + threadIdx.x;
  if (i < n) deg[i] = 1.0f;
}

// deg[dst[e]] += 1
__global__ void k_deg_count(const int* __restrict__ dst, float* __restrict__ deg, int e) {
  int i = blockIdx.x * blockDim.x + threadIdx.x;
  if (i < e) atomicAdd(&deg[dst[i]], 1.0f);
}

// dis[i] = deg[i]^{-1/2}   (deg >= 1 always, so rsqrt is safe)
__global__ void k_deg_rsqrt(float* __restrict__ deg, int n) {
  int i = blockIdx.x * blockDim.x + threadIdx.x;
  if (i < n) deg[i] = rsqrtf(deg[i]);
}

// H = act(X) @ W^T   with X: [n,64] row-major, W: [64,64] row-major.
// One wave handles a 16-row strip; 4 output N-tiles of 16 cols; K=64 in 16
// steps of 4 via V_WMMA_F32_16X16X4_F32 (full fp32 accuracy).
//
// ISA 7.12.2 fragment layouts (wave32):
//   A 16x4 : lane L, vgpr j -> (M = L%16, K = j + 2*(L/16))
//   B 4x16 : lane L, vgpr j -> (K = j + 2*(L/16), N = L%16)
//   D 16x16: lane L, vgpr v -> (M = v + 8*(L/16), N = L%16)
template <bool RELU_IN>
__global__ void k_gemm64_wmma(const float* __restrict__ X,
                              const float* __restrict__ W,
                              float* __restrict__ H, int nstrips) {
  int wave = (int)((blockIdx.x * blockDim.x + threadIdx.x) >> 5);
  int lane = threadIdx.x & 31;
  if (wave >= nstrips) return;          // whole-wave uniform: EXEC stays all-1s

  const int m0    = wave * 16;
  const int mlow  = lane & 15;          // M (for A) / N (for B,D)
  const int khalf = (lane >> 4) << 1;   // 0 or 2: K sub-offset for this half-wave

  // Preload all 16 A fragments for this row strip (32 VGPRs).
  v2f a[16];
  const float* xr = X + (size_t)(m0 + mlow) * 64 + khalf;
#pragma unroll
  for (int k = 0; k < 16; ++k) {
    float ax = xr[k * 4 + 0];
    float ay = xr[k * 4 + 1];
    if (RELU_IN) { ax = fmaxf(ax, 0.0f); ay = fmaxf(ay, 0.0f); }
    a[k].x = ax; a[k].y = ay;
  }

#pragma unroll
  for (int nt = 0; nt < 4; ++nt) {
    // B[k][n] = W[n][k]  (row-major W), so each fragment is a float2 from row n.
    const float* wr = W + (size_t)(nt * 16 + mlow) * 64 + khalf;
    v8f c = {};  // zero accumulator
#pragma unroll
    for (int k = 0; k < 16; ++k) {
      v2f b;
      b.x = wr[k * 4 + 0];
      b.y = wr[k * 4 + 1];
      c = __builtin_amdgcn_wmma_f32_16x16x4_f32(
          /*neg_a=*/false, a[k], /*neg_b=*/false, b,
          /*c_mod=*/(short)0, c, /*reuse_a=*/false, /*reuse_b=*/false);
    }
    float* o = H + (size_t)(m0 + 8 * (lane >> 4)) * 64 + nt * 16 + mlow;
#pragma unroll
    for (int v = 0; v < 8; ++v) o[(size_t)v * 64] = c[v];
  }
}

// out[i,:] = h[i,:] * dis[i]^2 + bias   (self-loop term + bias folded together)
__global__ void k_self_bias(const float* __restrict__ h, const float* __restrict__ dis,
                            const float* __restrict__ bias, float* __restrict__ out, int n) {
  int t = blockIdx.x * blockDim.x + threadIdx.x;
  if (t >= n * 16) return;
  int i = t >> 4;
  int f = (t & 15) << 2;
  float w = dis[i]; w *= w;
  float4 v = *(const float4*)(h + (size_t)i * 64 + f);
  float4 b = *(const float4*)(bias + f);
  float4 r;
  r.x = v.x * w + b.x; r.y = v.y * w + b.y;
  r.z = v.z * w + b.z; r.w = v.w * w + b.w;
  *(float4*)(out + (size_t)i * 64 + f) = r;
}

// out[dst[e],:] += h[src[e],:] * dis[src]*dis[dst]
// 16 threads per edge, float4 gather (coalesced), 4 f32 atomics each.
// h / out are L2-resident (25.6 MB each << 192 MB L2).
__global__ void k_edge_scatter(const int* __restrict__ src, const int* __restrict__ dst,
                               const float* __restrict__ h, const float* __restrict__ dis,
                               float* __restrict__ out, int e) {
  int t = blockIdx.x * blockDim.x + threadIdx.x;
  int ed = t >> 4;
  if (ed >= e) return;
  int f = (t & 15) << 2;
  int s = src[ed], d = dst[ed];
  float w = dis[s] * dis[d];
  float4 v = *(const float4*)(h + (size_t)s * 64 + f);
  float* o = out + (size_t)d * 64 + f;
  atomicAdd(o + 0, v.x * w);
  atomicAdd(o + 1, v.y * w);
  atomicAdd(o + 2, v.z * w);
  atomicAdd(o + 3, v.w * w);
}

extern "C" void kernel_launch(void* const* d_in, const int* in_sizes, int n_in,
                              void* d_out, int out_size, void* d_ws, size_t ws_size,
                              hipStream_t stream) {
  (void)n_in; (void)out_size; (void)ws_size;

  const float* x  = (const float*)d_in[0];   // [N,64]
  const int*   ei = (const int*)  d_in[1];   // [2,E] flat: src then dst
  const float* W1 = (const float*)d_in[2];   // [64,64]
  const float* b1 = (const float*)d_in[3];   // [64]
  const float* W2 = (const float*)d_in[4];   // [64,64]
  const float* b2 = (const float*)d_in[5];   // [64]
  float* out = (float*)d_out;                // [N,64]

  const int n = in_sizes[0] / 64;
  const int e = in_sizes[1] / 2;
  const int* src = ei;
  const int* dst = ei + e;

  // Workspace layout (floats): dis[n] | h[n*64] | agg[n*64]  (~52 MB)
  float* dis = (float*)d_ws;
  float* h   = dis + n;
  float* agg = h + (size_t)n * 64;

  const int nstrips = (n + 15) / 16;                 // 6250 for N=100000
  const int gemm_blocks = (nstrips + 7) / 8;         // 8 waves per 256-thr block
  const int nf_blocks   = (n * 16 + TB - 1) / TB;    // n*16 float4 threads
  const int ef_blocks   = (e * 16 + TB - 1) / TB;    // e*16 float4 threads

  // --- symmetric normalization: dis = (1 + in-degree)^{-1/2} ---
  k_deg_init <<<(n + TB - 1) / TB, TB, 0, stream>>>(dis, n);
  k_deg_count<<<(e + TB - 1) / TB, TB, 0, stream>>>(dst, dis, e);
  k_deg_rsqrt<<<(n + TB - 1) / TB, TB, 0, stream>>>(dis, n);

  // --- layer 1 ---
  k_gemm64_wmma<false><<<gemm_blocks, TB, 0, stream>>>(x, W1, h, nstrips);
  k_self_bias  <<<nf_blocks, TB, 0, stream>>>(h, dis, b1, agg, n);
  k_edge_scatter<<<ef_blocks, TB, 0, stream>>>(src, dst, h, dis, agg, e);

  // --- layer 2 (ReLU folded into the GEMM's A-operand load) ---
  k_gemm64_wmma<true><<<gemm_blocks, TB, 0, stream>>>(agg, W2, h, nstrips);
  k_self_bias  <<<nf_blocks, TB, 0, stream>>>(h, dis, b2, out, n);
  k_edge_scatter<<<ef_blocks, TB, 0, stream>>>(src, dst, h, dis, out, e);
}